// GCN_53558242181180
// MI455X (gfx1250) — compile-verified
//
#include <hip/hip_runtime.h>

// ---------------------------------------------------------------------------
// GCN 2-layer forward for MI455X (gfx1250, wave32).
// Layer-1 GEMM (128->16) on the WMMA pipe via v_wmma_f32_16x16x4_f32.
// Edge aggregation uses native f32 L2 atomics (agg buffers ~6.4MB -> fully
// resident in the 192MB L2). x is streamed with non-temporal loads so the
// one-pass 51MB stream doesn't evict the L2-resident aggregation data.
// ---------------------------------------------------------------------------

typedef __attribute__((ext_vector_type(2)))  float  v2f;
typedef __attribute__((ext_vector_type(8)))  float  v8f;

#define FIN  128
#define FMID 16

__device__ __forceinline__ void atomAddF(float* p, float v) {
  // native global_atomic_add_f32 (no CAS loop)
  unsafeAtomicAdd(p, v);
}

// ---- zero workspace ----
__global__ __launch_bounds__(256) void k_zero(unsigned* __restrict__ p, long total) {
  long i = (long)blockIdx.x * blockDim.x + threadIdx.x;
  long base = i * 4;
  if (base + 3 < total) {
    ((uint4*)p)[i] = make_uint4(0u, 0u, 0u, 0u);
  } else if (base < total) {
    for (long j = base; j < total; ++j) p[j] = 0u;
  }
}

// ---- degree of destination nodes (self-loop added as +1 later) ----
__global__ __launch_bounds__(256) void k_degree(const int* __restrict__ ei,
                                                unsigned* __restrict__ udeg, int E) {
  int e = blockIdx.x * blockDim.x + threadIdx.x;
  if (e >= E) return;
  atomicAdd(&udeg[ei[E + e]], 1u);
}

__global__ __launch_bounds__(256) void k_dinv(const unsigned* __restrict__ udeg,
                                              float* __restrict__ dinv, int N) {
  int i = blockIdx.x * blockDim.x + threadIdx.x;
  if (i >= N) return;
  dinv[i] = rsqrtf((float)(udeg[i] + 1u));  // +1 = self loop; deg>0 always
}

// ---- h1 = x @ W1 via WMMA: one wave computes one 16x16 output tile ----
__global__ __launch_bounds__(256) void k_gemm1_wmma(const float* __restrict__ x,
                                                    const float* __restrict__ W1,
                                                    float* __restrict__ h1, int N) {
  int wave = (int)((blockIdx.x * (unsigned)blockDim.x + threadIdx.x) >> 5);
  int lane = threadIdx.x & 31;
  int ntiles = (N + 15) >> 4;
  if (wave >= ntiles) return;         // wave-uniform exit: EXEC all-1s for WMMA
  int nb = wave << 4;
  int m  = lane & 15;
  int hi = lane >> 4;                 // half-wave select
  int row = nb + m; if (row > N - 1) row = N - 1;
  const float* xr = x + (size_t)row * FIN;
  v8f acc = {};

  // A 16x4 f32: lanes0-15 hold K={0,1}, lanes16-31 hold K={2,3} (VGPR0/1)
  // B 4x16  f32: same striping with N across lanes
  int kh = hi * 2;
  #pragma unroll 4
  for (int k = 0; k < FIN; k += 4) {
    v2f a, b;
    a.x = __builtin_nontemporal_load(&xr[k + kh]);      // single-use stream: TH=NT
    a.y = __builtin_nontemporal_load(&xr[k + kh + 1]);
    b.x = W1[(k + kh)     * FMID + m];                  // W1 tiny: keep cached
    b.y = W1[(k + kh + 1) * FMID + m];
    acc = __builtin_amdgcn_wmma_f32_16x16x4_f32(false, a, false, b,
                                                (short)0, acc, false, false);
  }

  // D layout: VGPR r -> (M=r, N=lane) for lanes0-15, (M=r+8, N=lane-16) for 16-31
  int rbase = hi * 8;
  if (nb + 16 <= N) {
    // wave-uniform full-tile fast path: unpredicated stores, no exec juggling
    #pragma unroll
    for (int r = 0; r < 8; ++r)
      h1[(size_t)(nb + rbase + r) * FMID + m] = acc[r];
  } else {
    #pragma unroll
    for (int r = 0; r < 8; ++r) {
      int nrow = nb + rbase + r;
      if (nrow < N) h1[(size_t)nrow * FMID + m] = acc[r];
    }
  }
}

// ---- layer-1 edge scatter: agg1[dst] += h1[src] * dinv[src]*dinv[dst] ----
__global__ __launch_bounds__(256) void k_agg1(const int* __restrict__ ei,
                                              const float* __restrict__ h1,
                                              const float* __restrict__ dinv,
                                              float* __restrict__ agg1, int E) {
  int e = blockIdx.x * blockDim.x + threadIdx.x;
  if (e >= E) return;
  int s = ei[e];
  int d = ei[E + e];
  float norm = dinv[s] * dinv[d];
  const float4* hr = (const float4*)(h1 + (size_t)s * FMID);
  float* ar = agg1 + (size_t)d * FMID;
  #pragma unroll
  for (int q = 0; q < 4; ++q) {
    float4 v = hr[q];
    atomAddF(ar + 4 * q + 0, v.x * norm);
    atomAddF(ar + 4 * q + 1, v.y * norm);
    atomAddF(ar + 4 * q + 2, v.z * norm);
    atomAddF(ar + 4 * q + 3, v.w * norm);
  }
}

// ---- out1 = agg1 + selfloop + b1; h2 = relu(out1) @ W2 (16->2, VALU) ----
__global__ __launch_bounds__(256) void k_layer2_in(const float* __restrict__ agg1,
                                                   const float* __restrict__ h1,
                                                   const float* __restrict__ dinv,
                                                   const float* __restrict__ b1,
                                                   const float* __restrict__ W2,
                                                   float* __restrict__ h2, int N) {
  int i = blockIdx.x * blockDim.x + threadIdx.x;
  if (i >= N) return;
  float d2 = dinv[i] * dinv[i];
  float a0 = 0.f, a1 = 0.f;
  #pragma unroll
  for (int k = 0; k < FMID; ++k) {
    float t = agg1[(size_t)i * FMID + k] + h1[(size_t)i * FMID + k] * d2 + b1[k];
    t = fmaxf(t, 0.f);
    a0 = fmaf(t, W2[k * 2 + 0], a0);
    a1 = fmaf(t, W2[k * 2 + 1], a1);
  }
  h2[(size_t)i * 2 + 0] = a0;
  h2[(size_t)i * 2 + 1] = a1;
}

// ---- layer-2 edge scatter ----
__global__ __launch_bounds__(256) void k_agg2(const int* __restrict__ ei,
                                              const float* __restrict__ h2,
                                              const float* __restrict__ dinv,
                                              float* __restrict__ agg2, int E) {
  int e = blockIdx.x * blockDim.x + threadIdx.x;
  if (e >= E) return;
  int s = ei[e];
  int d = ei[E + e];
  float norm = dinv[s] * dinv[d];
  float2 v = *(const float2*)(h2 + (size_t)s * 2);
  atomAddF(agg2 + (size_t)d * 2 + 0, v.x * norm);
  atomAddF(agg2 + (size_t)d * 2 + 1, v.y * norm);
}

// ---- out = agg2 + selfloop + b2 ----
__global__ __launch_bounds__(256) void k_final(const float* __restrict__ agg2,
                                               const float* __restrict__ h2,
                                               const float* __restrict__ dinv,
                                               const float* __restrict__ b2,
                                               float* __restrict__ out, int N) {
  int i = blockIdx.x * blockDim.x + threadIdx.x;
  if (i >= N) return;
  float d2 = dinv[i] * dinv[i];
  out[(size_t)i * 2 + 0] = agg2[(size_t)i * 2 + 0] + h2[(size_t)i * 2 + 0] * d2 + b2[0];
  out[(size_t)i * 2 + 1] = agg2[(size_t)i * 2 + 1] + h2[(size_t)i * 2 + 1] * d2 + b2[1];
}

extern "C" void kernel_launch(void* const* d_in, const int* in_sizes, int n_in,
                              void* d_out, int out_size, void* d_ws, size_t ws_size,
                              hipStream_t stream) {
  const float* x  = (const float*)d_in[0];
  const int*   ei = (const int*)d_in[1];    // edge_index [2, E]
  const float* W1 = (const float*)d_in[2];
  const float* b1 = (const float*)d_in[3];
  const float* W2 = (const float*)d_in[4];
  const float* b2 = (const float*)d_in[5];
  float* out = (float*)d_out;

  const int N = in_sizes[0] / FIN;   // 100000
  const int E = in_sizes[1] / 2;     // 3200000

  // workspace carve-up (38*N floats ~ 15.2 MB)
  char* ws = (char*)d_ws;
  unsigned* udeg = (unsigned*)ws;            ws += (size_t)N * sizeof(unsigned);
  float* dinv    = (float*)ws;               ws += (size_t)N * sizeof(float);
  float* h1      = (float*)ws;               ws += (size_t)N * FMID * sizeof(float);
  float* agg1    = (float*)ws;               ws += (size_t)N * FMID * sizeof(float);
  float* h2      = (float*)ws;               ws += (size_t)N * 2 * sizeof(float);
  float* agg2    = (float*)ws;               ws += (size_t)N * 2 * sizeof(float);

  const int TPB = 256;
  long zero_total = (long)N * 38;                     // whole carved region
  int gz = (int)(((zero_total + 3) / 4 + TPB - 1) / TPB);
  int ge = (E + TPB - 1) / TPB;
  int gn = (N + TPB - 1) / TPB;
  int ntiles = (N + 15) / 16;
  int gg = (ntiles * 32 + TPB - 1) / TPB;             // one wave (32 thr) per tile

  k_zero      <<<gz, TPB, 0, stream>>>((unsigned*)d_ws, zero_total);
  k_degree    <<<ge, TPB, 0, stream>>>(ei, udeg, E);
  k_dinv      <<<gn, TPB, 0, stream>>>(udeg, dinv, N);
  k_gemm1_wmma<<<gg, TPB, 0, stream>>>(x, W1, h1, N);
  k_agg1      <<<ge, TPB, 0, stream>>>(ei, h1, dinv, agg1, E);
  k_layer2_in <<<gn, TPB, 0, stream>>>(agg1, h1, dinv, b1, W2, h2, N);
  k_agg2      <<<ge, TPB, 0, stream>>>(ei, h2, dinv, agg2, E);
  k_final     <<<gn, TPB, 0, stream>>>(agg2, h2, dinv, b2, out, N);
}